// FeynmanIntegrand_89472758710336
// MI455X (gfx1250) — compile-verified
//
#include <hip/hip_runtime.h>
#include <hip/hip_bf16.h>

typedef __attribute__((ext_vector_type(2))) float v2f;
typedef __attribute__((ext_vector_type(8))) float v8f;

// ---------------- physics constants (constexpr-folded) ----------------
constexpr double D_PI    = 3.14159265358979323846;
constexpr double D_KF    = 0.9595791;              // (9*pi/4)^(1/3)/2
constexpr double D_EF    = D_KF * D_KF;            // KF^2/(2*ME), ME=0.5
constexpr double D_BETA  = 10.0 / D_EF;
constexpr double D_MU    = 0.99177533 * D_EF;
constexpr double D_MAXK  = 10.0 * D_KF;
constexpr double D_MASS2 = 0.5;
constexpr double D_BOSE  = 2.0 * 4.0 * D_PI;       // E0^2 / EPS0 = 2*(4*pi)
constexpr double D_2PI   = 2.0 * D_PI;
constexpr double D_T     = D_MAXK * 2.0 * D_PI * D_PI;
constexpr double D_2PI2  = D_2PI * D_2PI;
constexpr double D_2PI4  = D_2PI2 * D_2PI2;
constexpr double D_2PI9  = D_2PI4 * D_2PI4 * D_2PI;
constexpr double D_SCALE = (D_T * D_T * D_T) * (D_BETA * D_BETA) / D_2PI9;
constexpr double D_PIOB  = D_PI / D_BETA;

#define F_PI     ((float)D_PI)
#define F_2PI    ((float)D_2PI)
#define F_KF     ((float)D_KF)
#define F_BETA   ((float)D_BETA)
#define F_MU     ((float)D_MU)
#define F_MAXK   ((float)D_MAXK)
#define F_MASS2  ((float)D_MASS2)
#define F_BOSE   ((float)D_BOSE)
#define F_SCALE  ((float)D_SCALE)
#define F_PIOB   ((float)D_PIOB)

// ---------------- d_ws layout (32-bit word offsets) ----------------
// WS_PERM: int[3*128]
// WS_LEAF: float[128*16], one 64B record per leaf:
//   [0..3]  b0,b1,b2,b3   gathered basis column (loopBasis[i][momIdx[l]])
//   [4]     d0 = ((to==1)-(ti==1))*BETA     [5] d1 = ((to==2)-(ti==2))*BETA
//   [6]     ge1 = (lo1>=1)                  [7] ge2 = (lo1>=2)
//   [8]     fsel = (lft==1)                 [9] bsel = (lft==2)
//   [10]    dk2  = 2*dk                     [11] odk = 1-dk
//   [12]    m2dk = -2*(lo1+1)*dk            [13] cterm = (!fsel&&!bsel)*lv0
//   [14..15] spare
#define WS_PERM  0
#define WS_LEAF  384

// ---------------- MT19937 (numpy legacy RandomState) ----------------
struct MTState { unsigned mt[624]; int idx; };

__device__ inline void mt_seed(MTState& s, unsigned seed) {
    s.mt[0] = seed;
    for (int i = 1; i < 624; ++i)
        s.mt[i] = 1812433253u * (s.mt[i-1] ^ (s.mt[i-1] >> 30)) + (unsigned)i;
    s.idx = 624;
}
__device__ inline unsigned mt_next(MTState& s) {
    if (s.idx >= 624) {
        for (int i = 0; i < 624; ++i) {
            unsigned y = (s.mt[i] & 0x80000000u) | (s.mt[(i+1) % 624] & 0x7fffffffu);
            unsigned v = s.mt[(i+397) % 624] ^ (y >> 1);
            if (y & 1u) v ^= 2567483615u;
            s.mt[i] = v;
        }
        s.idx = 0;
    }
    unsigned y = s.mt[s.idx++];
    y ^= y >> 11;
    y ^= (y << 7)  & 2636928640u;
    y ^= (y << 15) & 4022730752u;
    y ^= y >> 18;
    return y;
}
__device__ inline unsigned mt_interval(MTState& s, unsigned mx) { // [0, mx]
    if (mx == 0u) return 0u;
    unsigned mask = mx;
    mask |= mask >> 1;  mask |= mask >> 2;  mask |= mask >> 4;
    mask |= mask >> 8;  mask |= mask >> 16;
    unsigned v;
    while ((v = (mt_next(s) & mask)) > mx) {}
    return v;
}

// ---------------- init: PERMS + packed branchless per-leaf constants ----
__global__ __launch_bounds__(128) void feyn_init(
    const int* __restrict__ lftype, const int* __restrict__ lforders,
    const int* __restrict__ taui,   const int* __restrict__ tauo,
    const int* __restrict__ momidx, const float* __restrict__ loopBasis,
    const float* __restrict__ lv0,  int* __restrict__ wsI, float* __restrict__ wsF)
{
    const int l = threadIdx.x;
    if (l < 128) {
        float* r = wsF + WS_LEAF + l * 16;
        const int mi = momidx[l];
        #pragma unroll
        for (int i = 0; i < 4; ++i) r[i] = loopBasis[i * 128 + mi];

        const int   ti = taui[l], to = tauo[l];
        const float so0 = (to == 1) ? 1.0f : 0.0f, so1 = (to == 2) ? 1.0f : 0.0f;
        const float si0 = (ti == 1) ? 1.0f : 0.0f, si1 = (ti == 2) ? 1.0f : 0.0f;
        r[4] = (so0 - si0) * F_BETA;
        r[5] = (so1 - si1) * F_BETA;

        const int lo1 = lforders[1 * 128 + l];
        r[6] = (lo1 >= 1) ? 1.0f : 0.0f;
        r[7] = (lo1 >= 2) ? 1.0f : 0.0f;

        const int   lt = lftype[l];
        const float dk = (lforders[2 * 128 + l] == 1) ? 1.0f : 0.0f;
        r[8]  = (lt == 1) ? 1.0f : 0.0f;
        r[9]  = (lt == 2) ? 1.0f : 0.0f;
        r[10] = 2.0f * dk;
        r[11] = 1.0f - dk;
        r[12] = -2.0f * ((float)lo1 + 1.0f) * dk;
        r[13] = (lt != 1 && lt != 2) ? lv0[l] : 0.0f;
        r[14] = 0.0f;
        r[15] = 0.0f;
    }
    if (l == 0) {
        MTState s;
        mt_seed(s, 0u);                       // np.random.RandomState(0)
        for (int o = 0; o < 3; ++o) {
            int a[128];
            for (int i = 0; i < 128; ++i) a[i] = i;
            for (int i = 127; i > 0; --i) {   // legacy shuffle: j in [0, i]
                const int j = (int)mt_interval(s, (unsigned)i);
                const int t = a[i]; a[i] = a[j]; a[j] = t;
            }
            for (int i = 0; i < 128; ++i) wsI[WS_PERM + o * 128 + i] = a[i];
        }
    }
}

// ---------------- main kernel ----------------
// block = 128 threads (4 waves), 64 batch elements / block.
// Each wave owns 16 batch rows; loops = p(16x4) @ basisCols(4x16) per leaf
// tile via V_WMMA_F32_16X16X4_F32, one per spatial dim per tile (24 total).
__global__ __launch_bounds__(128) void feyn_main(
    const float* __restrict__ var, const int* __restrict__ wsI,
    const float* __restrict__ wsF, float* __restrict__ out)
{
    __shared__ float lv[64 * 129];   // leaf values, padded stride 129
    __shared__ float v0s[64], v1s[64];
    __shared__ int   permS[384];

    const int tid   = threadIdx.x;
    const int bbase = blockIdx.x * 64;

    if (tid < 64) {
        v0s[tid] = var[(size_t)(bbase + tid) * 11 + 0];
        v1s[tid] = var[(size_t)(bbase + tid) * 11 + 1];
    }
    for (int i = tid; i < 384; i += 128) permS[i] = wsI[WS_PERM + i];
    __syncthreads();

    const int lane   = tid & 31;
    const int wave   = tid >> 5;
    const int hi     = lane >> 4;        // upper half of wave
    const int lane16 = lane & 15;
    const float fhi  = (float)hi;        // for branchless half-selects

    // ---- build A matrices: row M = lane16 (this wave's batch row) ----
    const float* v = var + (size_t)(bbase + wave * 16 + lane16) * 11;
    float p[3][4];
    p[0][0] = F_KF; p[1][0] = 0.0f; p[2][0] = 0.0f;
    #pragma unroll
    for (int l = 0; l < 3; ++l) {
        const float pr = v[2 + l] * F_MAXK;
        const float th = v[5 + l] * F_PI;
        const float ph = v[8 + l] * F_2PI;
        float st, ct, sp, cp;
        __sincosf(th, &st, &ct);
        __sincosf(ph, &sp, &cp);
        p[0][l + 1] = pr * st * cp;
        p[1][l + 1] = pr * st * sp;
        p[2][l + 1] = pr * ct;
    }
    v2f A[3];
    #pragma unroll
    for (int d = 0; d < 3; ++d) {        // lanes 0-15: K=0,1 ; lanes 16-31: K=2,3
        A[d].x = hi ? p[d][2] : p[d][0];
        A[d].y = hi ? p[d][3] : p[d][1];
    }

    // time vars for the 8 D-rows (M = e + 8*hi) this lane produces
    float tv0[8], tv1[8];
    #pragma unroll
    for (int e = 0; e < 8; ++e) {
        const int m = wave * 16 + hi * 8 + e;
        tv0[e] = v0s[m];
        tv1[e] = v1s[m];
    }

    // ---- leaf tiles: 8 tiles x 16 leaves; 3 WMMAs each (kx,ky,kz) ----
    #pragma unroll
    for (int t = 0; t < 8; ++t) {
        const int n = t * 16 + lane16;   // leaf column (N = lane16)
        const float4* lc = (const float4*)(wsF + WS_LEAF + n * 16);
        const float4 b4 = lc[0];
        const float4 c1 = lc[1];         // d0, d1, ge1, ge2
        const float4 c2 = lc[2];         // fsel, bsel, dk2, odk
        const float4 c3 = lc[3];         // m2dk, cterm, -, -

        v2f B;
        B.x = hi ? b4.z : b4.x;          // lanes 0-15: K=0,1 ; 16-31: K=2,3
        B.y = hi ? b4.w : b4.y;

        v8f c = {};
        v8f dx = __builtin_amdgcn_wmma_f32_16x16x4_f32(false, A[0], false, B, (short)0, c, false, false);
        v8f dy = __builtin_amdgcn_wmma_f32_16x16x4_f32(false, A[1], false, B, (short)0, c, false, false);
        v8f dz = __builtin_amdgcn_wmma_f32_16x16x4_f32(false, A[2], false, B, (short)0, c, false, false);

        #pragma unroll
        for (int e = 0; e < 8; ++e) {
            const float kx = dx[e], ky = dy[e], kz = dz[e];
            const float kq2 = kx * kx + ky * ky + kz * kz;

            // branchless tau: (v_o - v_i) * BETA
            const float tau  = c1.x * tv0[e] + c1.y * tv1[e];
            const float disp = kq2 - F_MU;            // /(2*ME) == *1

            const float sign = (tau > 0.0f) ? 1.0f : -1.0f;
            const float adjP = (disp > 0.0f) ? 0.0f    : F_BETA;  // tau>0 branch
            const float adjN = (disp > 0.0f) ? -F_BETA : 0.0f;    // tau<=0 branch
            const float a    = -tau + ((tau > 0.0f) ? adjP : adjN);
            const float bb   = (disp > 0.0f) ? -F_BETA : F_BETA;

            const float eA = __expf(disp * a);
            const float eB = __expf(disp * bb);
            float fermi = sign * eA * __builtin_amdgcn_rcpf(1.0f + eB);
            const float adf = kx * b4.x;
            fermi *= adf * c2.z + c2.w;               // adf*2dk + (1-dk)

            const float invK = __builtin_amdgcn_rcpf(kq2 + F_MASS2);
            const float m2i  = F_MASS2 * invK;
            const float m2im1 = m2i - 1.0f;
            const float powp = (1.0f + c1.z * m2im1) * (1.0f + c1.w * m2im1);
            float bose = F_BOSE * invK * powp;
            bose *= adf * invK * c3.x + c2.w;         // adf*invK*(-2)(lo1+1)dk + (1-dk)

            const float lvv = c2.x * fermi + c2.y * bose + c3.y;
            const int m = wave * 16 + hi * 8 + e;     // D-row -> local batch
            lv[m * 129 + n] = lvv;
        }
    }
    __syncthreads();

    // ---- stage B: per-batch permuted product-sum + scale + phase ----
    if (tid < 64) {
        const int b = bbase + tid;
        const float* vb = var + (size_t)b * 11;
        float factor = 1.0f;
        #pragma unroll
        for (int l = 0; l < 3; ++l) {
            const float pr = vb[2 + l] * F_MAXK;
            const float th = vb[5 + l] * F_PI;
            factor *= pr * pr * __sinf(th);
        }
        const float scale = factor * F_SCALE;
        const float* lvr = lv + tid * 129;
        float acc = 0.0f;
        #pragma unroll
        for (int o = 0; o < 3; ++o) {
            float s = 0.0f;
            for (int g = 0; g < 32; ++g) {
                float prod = 1.0f;
                #pragma unroll
                for (int k = 0; k < 4; ++k)
                    prod *= lvr[permS[o * 128 + g * 4 + k]];
                s += prod;
            }
            float phase;
            if (o == 0) phase = 1.0f;
            else        phase = __cosf(F_PIOB * vb[o - 1]);
            acc += s * phase;
        }
        out[b] = acc * scale;
    }
}

extern "C" void kernel_launch(void* const* d_in, const int* in_sizes, int n_in,
                              void* d_out, int out_size, void* d_ws, size_t ws_size,
                              hipStream_t stream) {
    const float* var       = (const float*)d_in[0];
    // d_in[1] = root (zeros, unused by the reference math)
    const int*   lftype    = (const int*)d_in[2];
    const int*   lforders  = (const int*)d_in[3];
    const int*   taui      = (const int*)d_in[4];
    const int*   tauo      = (const int*)d_in[5];
    const int*   momidx    = (const int*)d_in[6];
    const float* loopBasis = (const float*)d_in[7];
    const float* lv0       = (const float*)d_in[8];
    float*       out       = (float*)d_out;

    const int batch = in_sizes[0] / 11;

    feyn_init<<<1, 128, 0, stream>>>(lftype, lforders, taui, tauo, momidx,
                                     loopBasis, lv0, (int*)d_ws, (float*)d_ws);
    feyn_main<<<batch / 64, 128, 0, stream>>>(var, (const int*)d_ws,
                                              (const float*)d_ws, out);
}